// Net_57200374448180
// MI455X (gfx1250) — compile-verified
//
#include <hip/hip_runtime.h>
#include <hip/hip_bf16.h>

#define N_NODES 40000
#define N_EDGES 640000
#define D_IN    128
#define HC      128   // H*C
#define C_OUT   64
#define WT_STRIDE 132 // LDS row stride (floats): 132 % 64 == 4 -> conflict-free b64 reads

typedef float v2f __attribute__((ext_vector_type(2)));
typedef float v4f __attribute__((ext_vector_type(4)));
typedef float v8f __attribute__((ext_vector_type(8)));

__device__ __forceinline__ void atomic_max_float(float* addr, float val) {
    // signed-max for non-negative floats, unsigned-min for negative floats;
    // correct for mixed signs with -inf (0xFF800000) initialization.
    if (val >= 0.0f) atomicMax((int*)addr, __float_as_int(val));
    else             atomicMin((unsigned int*)addr, __float_as_uint(val));
}

// ---------------------------------------------------------------- init
__global__ void k_init(float* __restrict__ amax, float* __restrict__ denom,
                       float* __restrict__ agg) {
    const int i = blockIdx.x * blockDim.x + threadIdx.x;
    if (i < N_NODES * HC) agg[i] = 0.0f;
    if (i < N_NODES * 2) { amax[i] = -__builtin_inff(); denom[i] = 0.0f; }
}

// ---------------------------------------------------------------- GEMM
// out[M,128] = A[M,128] @ W[128,128] + bias[128]
// One wave per 16-row tile, 8 waves/block. W is staged TRANSPOSED in LDS
// so each B fragment (two consecutive K values of one column) is a single
// pair-aligned ds_load_b64 feeding v_wmma_f32_16x16x4_f32 directly.
__global__ void __launch_bounds__(256)
k_gemm128(const float* __restrict__ A, const float* __restrict__ W,
          const float* __restrict__ bias, float* __restrict__ out, int mtiles) {
    __shared__ float ldsWT[HC * WT_STRIDE];    // W^T [n][k], padded rows
    const int tid = threadIdx.x;
    // stage + transpose: ldsWT[n*WT_STRIDE + k] = W[k*HC + n]
    #pragma unroll
    for (int i = 0; i < (D_IN * HC) / 256; ++i) {
        const int idx = tid + i * 256;
        const int k = idx >> 7;
        const int n = idx & 127;
        ldsWT[n * WT_STRIDE + k] = W[idx];
    }
    __syncthreads();

    const int wave = tid >> 5;
    const int lane = tid & 31;
    const int half = lane >> 4;                // K pair select: +0 or +2
    const int l16  = lane & 15;                // M row (A) / N col (B,C,D)
    const int mtile = blockIdx.x * 8 + wave;
    if (mtile >= mtiles) return;

    const float* arow = A + (size_t)(mtile * 16 + l16) * D_IN + 2 * half;

    // 8 accumulators (one per 16-col tile), bias-initialized
    v8f acc[8];
    #pragma unroll
    for (int nt = 0; nt < 8; ++nt) {
        const float bv = bias[nt * 16 + l16];
        #pragma unroll
        for (int v = 0; v < 8; ++v) acc[nt][v] = bv;
    }

    // K loop: 32 chunks of 4; per chunk 1 A-frag load + 8 B-frag loads + 8 wmma
    #pragma unroll
    for (int c = 0; c < 32; ++c) {
        const v2f a = *(const v2f*)(arow + 4 * c);
        const int kk = 4 * c + 2 * half;
        #pragma unroll
        for (int nt = 0; nt < 8; ++nt) {
            const v2f b = *(const v2f*)(ldsWT + (size_t)(nt * 16 + l16) * WT_STRIDE + kk);
            acc[nt] = __builtin_amdgcn_wmma_f32_16x16x4_f32(
                false, a, false, b, (short)0, acc[nt], false, false);
        }
    }

    float* orow = out + (size_t)(mtile * 16) * HC;
    #pragma unroll
    for (int nt = 0; nt < 8; ++nt) {
        const int ncol = nt * 16 + l16;
        #pragma unroll
        for (int v = 0; v < 8; ++v)
            orow[(size_t)(v + 8 * half) * HC + ncol] = acc[nt][v];
    }
}

// ------------------------------------------------- alpha + segment max
// wave per edge: alpha[e,h] = leaky_relu( sum_c (le+xm[src])*att, 0.2 )
__global__ void __launch_bounds__(256)
k_alpha(const float* __restrict__ le, const float* __restrict__ xm,
        const float* __restrict__ att, const int* __restrict__ src,
        const int* __restrict__ dst, float* __restrict__ alpha,
        float* __restrict__ amax) {
    const int e = (blockIdx.x * blockDim.x + threadIdx.x) >> 5;
    const int lane = threadIdx.x & 31;
    if (e >= N_EDGES) return;
    const int s = src[e];
    const float* lrow = le + (size_t)e * HC;
    const float* xrow = xm + (size_t)s * HC;
    // cols lane, lane+32 -> head 0 ; cols lane+64, lane+96 -> head 1
    float p0 = (lrow[lane]      + xrow[lane])      * att[lane]
             + (lrow[lane + 32] + xrow[lane + 32]) * att[lane + 32];
    float p1 = (lrow[lane + 64] + xrow[lane + 64]) * att[lane + 64]
             + (lrow[lane + 96] + xrow[lane + 96]) * att[lane + 96];
    #pragma unroll
    for (int m = 16; m >= 1; m >>= 1) {
        p0 += __shfl_xor(p0, m, 32);
        p1 += __shfl_xor(p1, m, 32);
    }
    if (lane == 0) {
        const float a0 = p0 > 0.0f ? p0 : 0.2f * p0;
        const float a1 = p1 > 0.0f ? p1 : 0.2f * p1;
        alpha[2 * e]     = a0;
        alpha[2 * e + 1] = a1;
        const int d = dst[e];
        atomic_max_float(amax + 2 * d,     a0);
        atomic_max_float(amax + 2 * d + 1, a1);
    }
}

// --------------------------------------------- exp + segment sum (denom)
__global__ void k_softmax_num(float* __restrict__ alpha_ex,
                              const float* __restrict__ amax,
                              const int* __restrict__ dst,
                              float* __restrict__ denom) {
    const int t = blockIdx.x * blockDim.x + threadIdx.x;
    if (t >= N_EDGES * 2) return;
    const int e = t >> 1, h = t & 1;
    const int d = dst[e];
    const float ex = __expf(alpha_ex[t] - amax[2 * d + h]);
    alpha_ex[t] = ex;                 // in-place alpha -> ex
    atomicAdd(denom + 2 * d + h, ex);
}

// ------------------------------------------- weighted scatter aggregation
// wave per edge: agg[dst, col] += (le + xm[src]) * a_head(col)
__global__ void __launch_bounds__(256)
k_aggregate(const float* __restrict__ le, const float* __restrict__ xm,
            const float* __restrict__ ex, const float* __restrict__ denom,
            const int* __restrict__ src, const int* __restrict__ dst,
            float* __restrict__ agg) {
    const int e = (blockIdx.x * blockDim.x + threadIdx.x) >> 5;
    const int lane = threadIdx.x & 31;
    if (e >= N_EDGES) return;
    const int s = src[e];
    const int d = dst[e];
    const float a0 = ex[2 * e]     / (denom[2 * d]     + 1e-16f);
    const float a1 = ex[2 * e + 1] / (denom[2 * d + 1] + 1e-16f);
    const float* lrow = le + (size_t)e * HC;
    const float* xrow = xm + (size_t)s * HC;
    float* grow = agg + (size_t)d * HC;
    atomicAdd(grow + lane,      (lrow[lane]      + xrow[lane])      * a0);
    atomicAdd(grow + lane + 32, (lrow[lane + 32] + xrow[lane + 32]) * a0);
    atomicAdd(grow + lane + 64, (lrow[lane + 64] + xrow[lane + 64]) * a1);
    atomicAdd(grow + lane + 96, (lrow[lane + 96] + xrow[lane + 96]) * a1);
}

// ------------------------------------- out = mean_heads(agg) + x@W_self+b
__global__ void k_out(const float* __restrict__ x, const float* __restrict__ Wself,
                      const float* __restrict__ bself, const float* __restrict__ agg,
                      float* __restrict__ out) {
    const int t = blockIdx.x * blockDim.x + threadIdx.x;
    if (t >= N_NODES * C_OUT) return;
    const int n = t >> 6;
    const int c = t & 63;
    const float* xrow = x + (size_t)n * D_IN;
    float acc = bself[c];
    #pragma unroll 8
    for (int k = 0; k < D_IN; ++k)
        acc = fmaf(xrow[k], Wself[(size_t)k * C_OUT + c], acc);
    const float* grow = agg + (size_t)n * HC;
    out[t] = 0.5f * (grow[c] + grow[c + 64]) + acc;
}

// ---------------------------------------------------------------- launch
extern "C" void kernel_launch(void* const* d_in, const int* in_sizes, int n_in,
                              void* d_out, int out_size, void* d_ws, size_t ws_size,
                              hipStream_t stream) {
    (void)in_sizes; (void)n_in; (void)out_size; (void)ws_size;
    const float* x         = (const float*)d_in[0];
    const float* edge_attr = (const float*)d_in[1];
    const float* W_msg     = (const float*)d_in[2];
    const float* b_msg     = (const float*)d_in[3];
    const float* W_edge    = (const float*)d_in[4];
    const float* b_edge    = (const float*)d_in[5];
    const float* W_self    = (const float*)d_in[6];
    const float* b_self    = (const float*)d_in[7];
    const float* att       = (const float*)d_in[8];
    const int*   eidx      = (const int*)d_in[9];
    const int* src = eidx;              // edge_index[0]
    const int* dst = eidx + N_EDGES;    // edge_index[1]

    float* out = (float*)d_out;                        // [N, 64]
    float* le  = out + (size_t)N_NODES * C_OUT;        // [E, 128]

    float* ws    = (float*)d_ws;
    float* xm    = ws;                                 // N*128 = x@W_msg+b_msg
    float* alpha = xm    + (size_t)N_NODES * HC;       // E*2 (alpha, then ex)
    float* amax  = alpha + (size_t)N_EDGES * 2;        // N*2
    float* denom = amax  + (size_t)N_NODES * 2;        // N*2
    float* agg   = denom + (size_t)N_NODES * 2;        // N*128

    k_init<<<(N_NODES * HC + 255) / 256, 256, 0, stream>>>(amax, denom, agg);
    k_gemm128<<<(N_NODES / 16 + 7) / 8, 256, 0, stream>>>(x, W_msg, b_msg, xm, N_NODES / 16);
    k_gemm128<<<(N_EDGES / 16 + 7) / 8, 256, 0, stream>>>(edge_attr, W_edge, b_edge, le, N_EDGES / 16);
    k_alpha<<<(N_EDGES * 32 + 255) / 256, 256, 0, stream>>>(le, xm, att, src, dst, alpha, amax);
    k_softmax_num<<<(N_EDGES * 2 + 255) / 256, 256, 0, stream>>>(alpha, amax, dst, denom);
    k_aggregate<<<(N_EDGES * 32 + 255) / 256, 256, 0, stream>>>(le, xm, alpha, denom, src, dst, agg);
    k_out<<<(N_NODES * C_OUT + 255) / 256, 256, 0, stream>>>(x, W_self, b_self, agg, out);
}